// CapsuleLayer_39075612459355
// MI455X (gfx1250) — compile-verified
//
#include <hip/hip_runtime.h>

// f32 WMMA vector types (wave32): A/B of 16x16x4 f32 = 64 elems / 32 lanes = 2 VGPRs
typedef __attribute__((ext_vector_type(2))) float v2f;
typedef __attribute__((ext_vector_type(8))) float v8f;

// Problem constants: B=16, T=64, R=256, C=16, D_OUT=16, D_IN(=O)=16
// Workspace layout (float offsets)
#define WS_WSUM 0        // Wsum[r][c][o]  : 256*16*16 = 65536
#define WS_USUM 65536    // usum[b][t]     : 16*64     = 1024
#define WS_BIJ  66560    // b_ij[b][r][c]  : 16*256*16 = 65536
#define WS_CIJ  132096   // c_ij[b][r][c]  : 65536
#define WS_S    197632   // S[b][c][o]     : 16*16*16  = 4096
#define WS_N2   201728   // ||S[b][c]||^2  : 256
#define WS_G    201984   // g[b][c]        : 256
// total 202240 floats = 808960 bytes

// ---------------------------------------------------------------------------
// Kernel 1: Wsum[r,c,o] = sum_d W[r,c,d,o]; usum[b,t] = sum_i ui[b,t,i];
//           zero b_ij.  516 blocks x 256 threads = 132096 work items.
// ---------------------------------------------------------------------------
__global__ void caps_precompute(const float* __restrict__ ui,
                                const float* __restrict__ W,
                                float* __restrict__ ws) {
    int gid = blockIdx.x * blockDim.x + threadIdx.x;
    if (gid < 65536) {
        // W flat: [r][c][d][o], strides r=4096, c=256, d=16, o=1
        int rc = gid >> 4, o = gid & 15;
        const float* p = W + rc * 256 + o;
        float s = 0.f;
#pragma unroll
        for (int d = 0; d < 16; ++d) s += p[d * 16];
        ws[WS_WSUM + gid] = s;
    } else if (gid < 66560) {
        int bt = gid - 65536;
        const float* p = ui + bt * 16;
        float s = 0.f;
#pragma unroll
        for (int i = 0; i < 16; ++i) s += p[i];
        ws[WS_USUM + bt] = s;
    } else {
        ws[WS_BIJ + (gid - 66560)] = 0.f;
    }
}

// ---------------------------------------------------------------------------
// Kernel 2: row-wise softmax over c (16) of b_ij -> c_ij.
//           4096 rows (b*256+r); 16 blocks x 256 threads.
// ---------------------------------------------------------------------------
__global__ void caps_softmax(float* __restrict__ ws) {
    int row = blockIdx.x * blockDim.x + threadIdx.x;  // [0, 4096)
    const float* b = ws + WS_BIJ + row * 16;
    float* c = ws + WS_CIJ + row * 16;
    float m = b[0];
#pragma unroll
    for (int j = 1; j < 16; ++j) m = fmaxf(m, b[j]);
    float e[16];
    float s = 0.f;
#pragma unroll
    for (int j = 0; j < 16; ++j) { e[j] = __expf(b[j] - m); s += e[j]; }
    float inv = 1.f / s;
#pragma unroll
    for (int j = 0; j < 16; ++j) c[j] = e[j] * inv;
}

// ---------------------------------------------------------------------------
// Kernel 3: S[b,c,o] = sum_r c_ij[b,r,c] * Wsum[r,c,o]  via WMMA.
//           One wave per c: GEMM M=16(b) N=16(o) K=256(r), 64 k-steps of 4.
//           16 blocks x 32 threads.
// ---------------------------------------------------------------------------
__global__ void caps_s_gemm(float* __restrict__ ws) {
    const int c = blockIdx.x;        // 0..15
    const int lane = threadIdx.x;    // 0..31 (wave32, EXEC all ones)
    const int m = lane & 15;         // A row (b) / B col (o)
    const int half = lane >> 4;
    const float* Cij = ws + WS_CIJ;  // [b][r][c]
    const float* Wsm = ws + WS_WSUM; // [r][c][o]
    v8f acc = {};
    for (int kk = 0; kk < 256; kk += 4) {
        int k0 = kk + 2 * half;      // ISA A layout: lanes 0-15 hold K={0,1}, 16-31 K={2,3}
        v2f a, b;
        a.x = Cij[(m * 256 + k0) * 16 + c];
        a.y = Cij[(m * 256 + k0 + 1) * 16 + c];
        b.x = Wsm[(k0 * 16 + c) * 16 + m];
        b.y = Wsm[((k0 + 1) * 16 + c) * 16 + m];
        acc = __builtin_amdgcn_wmma_f32_16x16x4_f32(
            false, a, false, b, (short)0, acc, false, false);
    }
    float* S = ws + WS_S;            // [b][c][o]
#pragma unroll
    for (int v = 0; v < 8; ++v) {
        int row = v + 8 * half;      // D row -> b
        S[(row * 16 + c) * 16 + m] = acc[v];  // D col -> o
    }
}

// ---------------------------------------------------------------------------
// Kernel 4: n2[b,c] = ||S[b,c,:]||^2 ;
//           g[b,c]  = sum_t scale(u^2*n2) * u^2 , u = usum[b,t].
//           1 block x 256 threads (one per (b,c)).
// ---------------------------------------------------------------------------
__global__ void caps_norm_g(float* __restrict__ ws) {
    int tid = threadIdx.x;
    int b = tid >> 4, c = tid & 15;
    const float* S = ws + WS_S + (b * 16 + c) * 16;
    float n2s = 0.f;
#pragma unroll
    for (int o = 0; o < 16; ++o) n2s += S[o] * S[o];
    const float* u = ws + WS_USUM + b * 64;
    float g = 0.f;
#pragma unroll 4
    for (int t = 0; t < 64; ++t) {
        float uv = u[t];
        float u2 = uv * uv;
        float n2 = u2 * n2s;
        float sc = n2 / ((1.f + n2) * sqrtf(n2 + 1e-9f));
        g += sc * u2;
    }
    ws[WS_N2 + tid] = n2s;
    ws[WS_G + tid] = g;
}

// ---------------------------------------------------------------------------
// Kernel 5: b_ij[b,r,c] += g[b,c] * sum_o Wsum[r,c,o]*S[b,c,o]   via WMMA.
//           Per (c, r-tile): GEMM M=16(r) N=16(b) K=16(o), 4 k-steps.
//           256 blocks x 32 threads.
// ---------------------------------------------------------------------------
__global__ void caps_b_update(float* __restrict__ ws) {
    const int c = blockIdx.x & 15;
    const int rt = blockIdx.x >> 4;  // r tile 0..15
    const int lane = threadIdx.x;
    const int m = lane & 15;         // A row (r within tile) / B col (b)
    const int half = lane >> 4;
    const float* Wsm = ws + WS_WSUM; // [r][c][o]
    const float* S = ws + WS_S;      // [b][c][o]
    const float* g = ws + WS_G;      // [b][c]
    float* bij = ws + WS_BIJ;        // [b][r][c]
    v8f acc = {};
#pragma unroll
    for (int kk = 0; kk < 16; kk += 4) {
        int k0 = kk + 2 * half;
        v2f a, b;
        a.x = Wsm[((rt * 16 + m) * 16 + c) * 16 + k0];
        a.y = Wsm[((rt * 16 + m) * 16 + c) * 16 + k0 + 1];
        b.x = S[(m * 16 + c) * 16 + k0];
        b.y = S[(m * 16 + c) * 16 + k0 + 1];
        acc = __builtin_amdgcn_wmma_f32_16x16x4_f32(
            false, a, false, b, (short)0, acc, false, false);
    }
#pragma unroll
    for (int v = 0; v < 8; ++v) {
        int r = rt * 16 + v + 8 * half;  // D row -> r
        int bb = m;                      // D col -> b
        bij[(bb * 256 + r) * 16 + c] += g[bb * 16 + c] * acc[v];
    }
}

// ---------------------------------------------------------------------------
// Kernel 6: v_j[b,t,c,o] = scale * usum[b,t] * S[b,c,o],
//           scale = n2/(1+n2)/sqrt(n2+eps), n2 = usum^2 * ||S[b,c]||^2.
//           1024 blocks x 256 threads = 262144 outputs.
// ---------------------------------------------------------------------------
__global__ void caps_vout(const float* __restrict__ ws, float* __restrict__ out) {
    int gid = blockIdx.x * blockDim.x + threadIdx.x;
    int o = gid & 15;
    int c = (gid >> 4) & 15;
    int t = (gid >> 8) & 63;
    int b = gid >> 14;
    float u = ws[WS_USUM + b * 64 + t];
    float n2 = u * u * ws[WS_N2 + b * 16 + c];
    float sc = n2 / ((1.f + n2) * sqrtf(n2 + 1e-9f));
    out[gid] = sc * u * ws[WS_S + (b * 16 + c) * 16 + o];
}

extern "C" void kernel_launch(void* const* d_in, const int* in_sizes, int n_in,
                              void* d_out, int out_size, void* d_ws, size_t ws_size,
                              hipStream_t stream) {
    const float* ui = (const float*)d_in[0];  // [16,64,16]
    const float* W  = (const float*)d_in[1];  // [1,256,16,16,16]
    float* out = (float*)d_out;               // [16,64,16,16]
    float* ws  = (float*)d_ws;                // >= 808960 bytes used

    caps_precompute<<<516, 256, 0, stream>>>(ui, W, ws);
    for (int it = 0; it < 3; ++it) {
        caps_softmax<<<16, 256, 0, stream>>>(ws);
        caps_s_gemm<<<16, 32, 0, stream>>>(ws);
        caps_norm_g<<<1, 256, 0, stream>>>(ws);
        if (it < 2) caps_b_update<<<256, 32, 0, stream>>>(ws);
    }
    caps_vout<<<1024, 256, 0, stream>>>(ws, out);
}